// FirstInteraction_69776038691501
// MI455X (gfx1250) — compile-verified
//
#include <hip/hip_runtime.h>
#include <hip/hip_bf16.h>

typedef __attribute__((ext_vector_type(2))) float v2f;
typedef __attribute__((ext_vector_type(8))) float v8f;

#define NR 8           // radial
#define ZPA 8          // Z floats per atom
#define FPA 24         // F floats per atom (8x3)

// ---------------------------------------------------------------------------
// Kernel 1: per-edge moments + scatter-add into Z[a,8] / F[a,8,3] workspace.
// 128 MB of accumulators fits in the 192 MB L2, so the f32 atomics stay
// on-chip (global_atomic_add_f32).
// ---------------------------------------------------------------------------
__global__ void fi_scatter_kernel(const float* __restrict__ dn,
                                  const float* __restrict__ h_s,
                                  const float* __restrict__ basis,
                                  const int* __restrict__ idx_i,
                                  float* __restrict__ Z,
                                  float* __restrict__ F,
                                  int n_edges) {
    int e = blockIdx.x * blockDim.x + threadIdx.x;
    if (e >= n_edges) return;

    const float4* hs4 = (const float4*)(h_s + (size_t)e * NR);
    const float4* bs4 = (const float4*)(basis + (size_t)e * NR);
    float4 h0 = hs4[0], h1 = hs4[1];
    float4 b0 = bs4[0], b1 = bs4[1];

    float zm[NR];
    zm[0] = h0.x * b0.x; zm[1] = h0.y * b0.y;
    zm[2] = h0.z * b0.z; zm[3] = h0.w * b0.w;
    zm[4] = h1.x * b1.x; zm[5] = h1.y * b1.y;
    zm[6] = h1.z * b1.z; zm[7] = h1.w * b1.w;

    float d0 = dn[(size_t)e * 3 + 0];
    float d1 = dn[(size_t)e * 3 + 1];
    float d2 = dn[(size_t)e * 3 + 2];

    int a = idx_i[e];
    float* zp = Z + (size_t)a * ZPA;
    float* fp = F + (size_t)a * FPA;

#pragma unroll
    for (int r = 0; r < NR; ++r) {
        unsafeAtomicAdd(&zp[r], zm[r]);
        unsafeAtomicAdd(&fp[r * 3 + 0], zm[r] * d0);
        unsafeAtomicAdd(&fp[r * 3 + 1], zm[r] * d1);
        unsafeAtomicAdd(&fp[r * 3 + 2], zm[r] * d2);
    }
}

// ---------------------------------------------------------------------------
// Kernel 2: per-atom contractions. One wave32 handles TWO atoms.
// h_s1 = F·F^T (8x3 x 3x8) for both atoms via ONE v_wmma_f32_16x16x4_f32:
//   A(16x4): rows 0-7 = F[a0] (K=3, pad K3=0), rows 8-15 = F[a1]
//   B(4x16) = A^T  -> with the ISA register layouts the A and B per-lane
//   fragments are bit-identical (same 12-byte row load).
//   D's two 8x8 diagonal blocks are h_s1[a0] and h_s1[a1].
// EXEC must be all-1s at the WMMA: out-of-range atoms are clamped on load
// and masked only at the stores.
// ---------------------------------------------------------------------------
__global__ void fi_contract_kernel(const float* __restrict__ Z,
                                   const float* __restrict__ F,
                                   float* __restrict__ out,
                                   int n_atoms) {
    const int lane = threadIdx.x & 31;
    const int wave_in_blk = threadIdx.x >> 5;
    const long long wave = (long long)blockIdx.x * (blockDim.x >> 5) + wave_in_blk;
    const long long base = wave * 2;               // uniform across the wave
    if (base >= n_atoms) return;                   // whole-wave exit only

    // ---- build the shared A/B fragment (16x4 f32 layout) ----
    const int idx16 = lane & 15;                   // M (rows) == N (cols) mapping
    long long aL = base + (idx16 >> 3);            // which atom this lane feeds
    if (aL >= n_atoms) aL = n_atoms - 1;           // clamp: keep EXEC all-1s
    const int r = idx16 & 7;
    const float* Fr = F + aL * FPA + r * 3;
    const float f0 = Fr[0], f1 = Fr[1], f2 = Fr[2];

    const bool hi = lane >= 16;                    // lanes 16-31 carry K=2,3
    v2f frag;
    frag.x = hi ? f2 : f0;
    frag.y = hi ? 0.0f : f1;

    v8f acc = {};
    // D = A x B + 0  -> v_wmma_f32_16x16x4_f32
    acc = __builtin_amdgcn_wmma_f32_16x16x4_f32(
        /*neg_a=*/false, frag, /*neg_b=*/false, frag,
        /*c_mod=*/(short)0, acc, /*reuse_a=*/false, /*reuse_b=*/false);

    // ---- store h_s_out = [Z(8) | h_s1(64)] for the diagonal blocks ----
    // D layout: VGPR v, lanes 0-15 -> D[v][lane]; lanes 16-31 -> D[v+8][lane-16].
    // atom0 block: rows 0-7, cols 0-7  -> lanes 0-7,  component v = row.
    // atom1 block: rows 8-15, cols 8-15 -> lanes 24-31, component v = row-8.
    long long aout = -1; int s = 0;
    if (lane < 8)        { aout = base;     s = lane; }
    else if (lane >= 24) { aout = base + 1; s = lane - 24; }
    if (aout >= 0 && aout < n_atoms) {
        float* o = out + aout * 72;
        o[s] = Z[aout * ZPA + s];                  // h_s0 column s
        float* o1 = o + 8 + s;                     // h_s1[:, s]
#pragma unroll
        for (int rr = 0; rr < 8; ++rr) o1[rr * 8] = acc[rr];
    }

    // ---- h_p[a][i][r][s] = Z[a][r] * F[a][s][i] : 192 f/atom, 384 f/wave ----
    // Lane-striped fully coalesced stores; operand re-reads hit L0 (2 atoms
    // of Z|F = 256 B).
    const long long hp_off = (long long)n_atoms * 72;
#pragma unroll
    for (int j = 0; j < 12; ++j) {
        int e = j * 32 + lane;                     // 0..383
        long long atom = base + (e >= 192 ? 1 : 0);
        if (atom < n_atoms) {
            int rem = e - (e >= 192 ? 192 : 0);    // 0..191
            int i  = rem >> 6;                     // spatial 0..2
            int rs = rem & 63;
            int rr = rs >> 3;
            int ss = rs & 7;
            float v = Z[atom * ZPA + rr] * F[atom * FPA + ss * 3 + i];
            out[hp_off + atom * 192 + rem] = v;
        }
    }
}

extern "C" void kernel_launch(void* const* d_in, const int* in_sizes, int n_in,
                              void* d_out, int out_size, void* d_ws, size_t ws_size,
                              hipStream_t stream) {
    const float* dn    = (const float*)d_in[0];
    const float* h_s   = (const float*)d_in[1];
    const float* basis = (const float*)d_in[2];
    const int*   idx_i = (const int*)d_in[3];

    const int n_edges = in_sizes[0] / 3;   // dn is [n_edges, 3]
    const int n_atoms = in_sizes[1] / NR;  // h_s is [n_atoms, 8]

    float* Z = (float*)d_ws;                        // [n_atoms, 8]
    float* F = Z + (size_t)n_atoms * ZPA;           // [n_atoms, 8, 3]
    const size_t acc_bytes = (size_t)n_atoms * (ZPA + FPA) * sizeof(float);

    // Zero the accumulators every call (scatter accumulates into them).
    hipMemsetAsync(d_ws, 0, acc_bytes, stream);

    {
        dim3 blk(256);
        dim3 grd((n_edges + 255) / 256);
        fi_scatter_kernel<<<grd, blk, 0, stream>>>(dn, h_s, basis, idx_i, Z, F, n_edges);
    }
    {
        const long long waves = ((long long)n_atoms + 1) / 2;   // 2 atoms / wave
        dim3 blk(256);                                          // 8 waves / block
        dim3 grd((unsigned)((waves + 7) / 8));
        fi_contract_kernel<<<grd, blk, 0, stream>>>(Z, F, (float*)d_out, n_atoms);
    }
}